// SimpleGNN_13219909337227
// MI455X (gfx1250) — compile-verified
//
#include <hip/hip_runtime.h>

#define NN 100000
#define NE 1600000
#define IND 128
#define HID 64
#define ROWTILES (NN / 16)   // 6250, exact

typedef __attribute__((ext_vector_type(2))) float v2f;
typedef __attribute__((ext_vector_type(8))) float v8f;

// ---------------- Input GEMM: H0 = relu(X @ Wi + bi) ------------------------
// block = 128 threads (4 wave32), each wave computes a 16x64 output tile.
__global__ __launch_bounds__(128) void gnn_gemm_in(
    const float* __restrict__ X, const float* __restrict__ W,
    const float* __restrict__ bias, float* __restrict__ OUT) {
  __shared__ float sW[IND * HID];  // 32 KB
  const int tid = threadIdx.x;
  for (int i = tid; i < IND * HID; i += 128) sW[i] = W[i];
  __syncthreads();

  const int wave = tid >> 5;
  const int lane = tid & 31;
  const int rowtile = blockIdx.x * 4 + wave;
  if (rowtile >= ROWTILES) return;  // wave-uniform: EXEC stays all-1s for WMMA

  const int m0 = rowtile * 16;
  const int lrow = lane & 15;          // M (for A) / N (for B,C)
  const int khalf = (lane >> 4) << 1;  // 0 or 2

  const float* xrow = X + (size_t)(m0 + lrow) * IND;
  v8f acc[4];
#pragma unroll
  for (int t = 0; t < 4; ++t) acc[t] = (v8f){0.f,0.f,0.f,0.f,0.f,0.f,0.f,0.f};

  for (int k = 0; k < IND; k += 4) {
    const int kb = k + khalf;
    v2f a;
    a.x = xrow[kb];
    a.y = xrow[kb + 1];
    const float* wp = sW + kb * HID + lrow;
#pragma unroll
    for (int nt = 0; nt < 4; ++nt) {
      v2f b;
      b.x = wp[nt * 16];
      b.y = wp[HID + nt * 16];
      acc[nt] = __builtin_amdgcn_wmma_f32_16x16x4_f32(
          false, a, false, b, (short)0, acc[nt], false, false);
    }
  }

  const int rowb = m0 + ((lane >> 4) << 3);
  float* op = OUT + (size_t)rowb * HID + lrow;
#pragma unroll
  for (int nt = 0; nt < 4; ++nt) {
    const float bb = bias[nt * 16 + lrow];
#pragma unroll
    for (int v = 0; v < 8; ++v)
      op[(size_t)v * HID + nt * 16] = fmaxf(acc[nt][v] + bb, 0.f);
  }
}

// ---------------- Layer GEMM: Hn = relu((Hp + M) @ W + b) -------------------
__global__ __launch_bounds__(128) void gnn_gemm_layer(
    const float* __restrict__ HP, const float* __restrict__ MSG,
    const float* __restrict__ W, const float* __restrict__ bias,
    float* __restrict__ OUT) {
  __shared__ float sW[HID * HID];  // 16 KB
  const int tid = threadIdx.x;
  for (int i = tid; i < HID * HID; i += 128) sW[i] = W[i];
  __syncthreads();

  const int wave = tid >> 5;
  const int lane = tid & 31;
  const int rowtile = blockIdx.x * 4 + wave;
  if (rowtile >= ROWTILES) return;

  const int m0 = rowtile * 16;
  const int lrow = lane & 15;
  const int khalf = (lane >> 4) << 1;

  const size_t roff = (size_t)(m0 + lrow) * HID;
  const float* hrow = HP + roff;
  const float* mrow = MSG + roff;
  v8f acc[4];
#pragma unroll
  for (int t = 0; t < 4; ++t) acc[t] = (v8f){0.f,0.f,0.f,0.f,0.f,0.f,0.f,0.f};

  for (int k = 0; k < HID; k += 4) {
    const int kb = k + khalf;
    v2f a;
    a.x = hrow[kb] + mrow[kb];
    a.y = hrow[kb + 1] + mrow[kb + 1];
    const float* wp = sW + kb * HID + lrow;
#pragma unroll
    for (int nt = 0; nt < 4; ++nt) {
      v2f b;
      b.x = wp[nt * 16];
      b.y = wp[HID + nt * 16];
      acc[nt] = __builtin_amdgcn_wmma_f32_16x16x4_f32(
          false, a, false, b, (short)0, acc[nt], false, false);
    }
  }

  const int rowb = m0 + ((lane >> 4) << 3);
  float* op = OUT + (size_t)rowb * HID + lrow;
#pragma unroll
  for (int nt = 0; nt < 4; ++nt) {
    const float bb = bias[nt * 16 + lrow];
#pragma unroll
    for (int v = 0; v < 8; ++v)
      op[(size_t)v * HID + nt * 16] = fmaxf(acc[nt][v] + bb, 0.f);
  }
}

// ---------------- Edge scatter: MSG[tgt] += H[src] * w ----------------------
// 16 threads per edge, float4 gather, native f32 atomic adds (no-return).
__global__ __launch_bounds__(256) void gnn_scatter(
    const int* __restrict__ EI, const float* __restrict__ EA,
    const float* __restrict__ H, float* __restrict__ MSG) {
  const int t = blockIdx.x * 256 + threadIdx.x;
  const int e = t >> 4;
  const int q = t & 15;
  if (e >= NE) return;
  const int s = EI[e];
  const int d = EI[NE + e];
  const float w = EA[e];
  const float4 hv = *((const float4*)(H + (size_t)s * HID) + q);
  float* mp = MSG + (size_t)d * HID + (size_t)q * 4;
  unsafeAtomicAdd(mp + 0, hv.x * w);
  unsafeAtomicAdd(mp + 1, hv.y * w);
  unsafeAtomicAdd(mp + 2, hv.z * w);
  unsafeAtomicAdd(mp + 3, hv.w * w);
}

// ---------------- Zero the message buffer -----------------------------------
__global__ __launch_bounds__(256) void gnn_zero(float4* __restrict__ p, int n4) {
  const int i = blockIdx.x * 256 + threadIdx.x;
  if (i < n4) p[i] = make_float4(0.f, 0.f, 0.f, 0.f);
}

extern "C" void kernel_launch(void* const* d_in, const int* in_sizes, int n_in,
                              void* d_out, int out_size, void* d_ws, size_t ws_size,
                              hipStream_t stream) {
  const float* x  = (const float*)d_in[0];
  const int*   ei = (const int*)d_in[1];
  const float* ea = (const float*)d_in[2];
  const float* Wi = (const float*)d_in[3];
  const float* bi = (const float*)d_in[4];
  const float* Ws[3] = {(const float*)d_in[5], (const float*)d_in[7], (const float*)d_in[9]};
  const float* bs[3] = {(const float*)d_in[6], (const float*)d_in[8], (const float*)d_in[10]};

  float* out = (float*)d_out;           // states: 4 x NN x HID, h_l lives in slice l
  float* msg = (float*)d_ws;            // NN x HID scratch (25.6 MB)
  const size_t NH = (size_t)NN * HID;

  const int gemm_blocks = (ROWTILES + 3) / 4;  // 4 waves/block, 1 rowtile each
  gnn_gemm_in<<<gemm_blocks, 128, 0, stream>>>(x, Wi, bi, out);

  const int n4 = (int)(NH / 4);
  for (int l = 0; l < 3; ++l) {
    gnn_zero<<<(n4 + 255) / 256, 256, 0, stream>>>((float4*)msg, n4);
    gnn_scatter<<<(NE * 16) / 256, 256, 0, stream>>>(ei, ea, out + (size_t)l * NH, msg);
    gnn_gemm_layer<<<gemm_blocks, 128, 0, stream>>>(out + (size_t)l * NH, msg,
                                                    Ws[l], bs[l], out + (size_t)(l + 1) * NH);
  }
}